// DetectionPostProcessor_30502857736330
// MI455X (gfx1250) — compile-verified
//
#include <hip/hip_runtime.h>
#include <hip/hip_bf16.h>

// ---------------- constants ----------------
#define SCORE_THRESH 0.05f
#define NMS_THRESH   0.5f
#define DETS_PER_IMG 300
#define TOPK_N       1000
#define NBINS        4096
#define CAND_CAP     4096
#define NEG_INF_F    (-__builtin_inff())

// ---------------- workspace layout (bytes) ----------------
#define OFF_HIST    0u          // 4096 u32
#define OFF_FLAGS   16384u      // 16 u32   [0]=thrBin [2]=candCount
#define OFF_CSCORE  16448u      // 4096 f32
#define OFF_CIDX    32832u      // 4096 u32
#define OFF_VALS    49216u      // 1024 f32
#define OFF_IDX     53312u      // 1024 i32
#define OFF_BOXK    57408u      // 1024*5 f32
#define OFF_LABK    77888u      // 1024 i32
#define OFF_AB      81984u      // 1024 float4 (16B aligned)
#define OFF_CORN    98368u      // 1024*8 f32
#define OFF_RAD     131136u     // 1024 f32
#define OFF_AREA    135232u     // 1024 f32
#define OFF_MASK    139328u     // 1024*32 u32 suppression bits
#define OFF_KEEP    270400u     // 32 u32
#define WS_TOTAL    270528u

typedef float v2f __attribute__((ext_vector_type(2)));
typedef float v8f __attribute__((ext_vector_type(8)));

// ---------------- K0: zero workspace ----------------
__global__ void k0_zero(unsigned* ws, int nwords) {
    for (int t = blockIdx.x * blockDim.x + threadIdx.x; t < nwords; t += gridDim.x * blockDim.x)
        ws[t] = 0u;
}

// ---------------- K1: histogram of scores > thresh ----------------
__global__ void k1_hist(const float* __restrict__ scores, int n, unsigned* __restrict__ hist) {
    __shared__ unsigned sh[NBINS];
    for (int t = threadIdx.x; t < NBINS; t += blockDim.x) sh[t] = 0u;
    __syncthreads();
    for (int g = blockIdx.x * blockDim.x + threadIdx.x; g < n; g += gridDim.x * blockDim.x) {
        float sv = scores[g];
        if (sv > SCORE_THRESH) {
            int b = (int)(sv * (float)NBINS);
            b = b < 0 ? 0 : (b > NBINS - 1 ? NBINS - 1 : b);
            atomicAdd(&sh[b], 1u);
        }
    }
    __syncthreads();
    for (int t = threadIdx.x; t < NBINS; t += blockDim.x)
        if (sh[t]) atomicAdd(&hist[t], sh[t]);
}

// ---------------- K2: scan from top to find threshold bin ----------------
__global__ void k2_scan(const unsigned* __restrict__ hist, unsigned* __restrict__ flags) {
    if (threadIdx.x == 0 && blockIdx.x == 0) {
        unsigned cum = 0; int thr = 0;
        for (int b = NBINS - 1; b >= 0; --b) {
            cum += hist[b];
            if (cum >= (unsigned)TOPK_N) { thr = b; break; }
        }
        flags[0] = (unsigned)thr;
    }
}

// ---------------- K3: compact candidates above threshold bin ----------------
__global__ void k3_compact(const float* __restrict__ scores, int n,
                           const unsigned* __restrict__ flags, unsigned* __restrict__ cnt,
                           float* __restrict__ cs, unsigned* __restrict__ ci) {
    int thr = (int)flags[0];
    for (int g = blockIdx.x * blockDim.x + threadIdx.x; g < n; g += gridDim.x * blockDim.x) {
        float sv = scores[g];
        if (sv > SCORE_THRESH) {
            int b = (int)(sv * (float)NBINS);
            b = b < 0 ? 0 : (b > NBINS - 1 ? NBINS - 1 : b);
            if (b >= thr) {
                unsigned p = atomicAdd(cnt, 1u);
                if (p < (unsigned)CAND_CAP) { cs[p] = sv; ci[p] = (unsigned)g; }
            }
        }
    }
}

// ---------------- K4: bitonic sort 4096 candidates desc, emit top-1024 ----------------
__global__ void k4_sort(const float* __restrict__ cs, const unsigned* __restrict__ ci,
                        const unsigned* __restrict__ flags,
                        float* __restrict__ vals, int* __restrict__ idxOut) {
    __shared__ float s[CAND_CAP];
    __shared__ int   id[CAND_CAP];
    int cnt = (int)flags[2];
    if (cnt > CAND_CAP) cnt = CAND_CAP;
    for (int t = threadIdx.x; t < CAND_CAP; t += blockDim.x) {
        if (t < cnt) { s[t] = cs[t]; id[t] = (int)ci[t]; }
        else         { s[t] = NEG_INF_F; id[t] = 0x7fffffff; }
    }
    __syncthreads();
    for (unsigned k = 2; k <= (unsigned)CAND_CAP; k <<= 1) {
        for (unsigned j = k >> 1; j > 0; j >>= 1) {
            for (unsigned t = threadIdx.x; t < (unsigned)CAND_CAP; t += blockDim.x) {
                unsigned ixj = t ^ j;
                if (ixj > t) {
                    bool up = ((t & k) == 0);
                    float sa = s[t], sb = s[ixj];
                    int   ia = id[t], ib = id[ixj];
                    bool before = (sa > sb) || (sa == sb && ia < ib);
                    if (up ? !before : before) {
                        s[t] = sb; id[t] = ib; s[ixj] = sa; id[ixj] = ia;
                    }
                }
            }
            __syncthreads();
        }
    }
    for (int t = threadIdx.x; t < 1024; t += blockDim.x) { vals[t] = s[t]; idxOut[t] = id[t]; }
}

// ---------------- K5: gather top-k boxes, derive geometry + GEMM fragments ----------------
__global__ void k5_gather(const float* __restrict__ boxes, const int* __restrict__ labels,
                          const float* __restrict__ vals, const int* __restrict__ idx,
                          float* __restrict__ boxes_k, int* __restrict__ labels_k,
                          float4* __restrict__ ab, float* __restrict__ corners,
                          float* __restrict__ radius, float* __restrict__ area) {
    int t = blockIdx.x * blockDim.x + threadIdx.x;
    if (t >= 1024) return;
    float cx = 0.f, cy = 0.f, w = 0.f, h = 0.f, a = 0.f;
    int lb = -1;
    if (t < TOPK_N) {
        float v = vals[t];
        if (v > -1e30f) {
            int j = idx[t];
            cx = boxes[j * 5 + 0]; cy = boxes[j * 5 + 1];
            w  = boxes[j * 5 + 2]; h  = boxes[j * 5 + 3];
            a  = boxes[j * 5 + 4];
            lb = labels[j];
        }
    }
    boxes_k[t * 5 + 0] = cx; boxes_k[t * 5 + 1] = cy;
    boxes_k[t * 5 + 2] = w;  boxes_k[t * 5 + 3] = h;  boxes_k[t * 5 + 4] = a;
    labels_k[t] = lb;
    float c = cosf(a), sn = sinf(a);
    float dx = 0.5f * w, dy = 0.5f * h;
    float ox[4] = { dx, -dx, -dx, dx };
    float oy[4] = { dy,  dy, -dy, -dy };
#pragma unroll
    for (int q = 0; q < 4; ++q) {
        corners[t * 8 + 2 * q + 0] = cx + ox[q] * c - oy[q] * sn;
        corners[t * 8 + 2 * q + 1] = cy + ox[q] * sn + oy[q] * c;
    }
    radius[t] = 0.5f * sqrtf(w * w + h * h) + 2.0f;  // +margin for gate conservatism
    area[t]   = w * h;
    float4 p; p.x = cx; p.y = cy; p.z = cx * cx + cy * cy; p.w = 1.0f;
    ab[t] = p;
}

// ---------------- exact convex intersection (Sutherland–Hodgman, CCW boxes) ----------------
__device__ float inter_area(const float* __restrict__ cA, const float* __restrict__ cB) {
    float px[16], py[16];
    int n = 4;
#pragma unroll
    for (int t = 0; t < 4; ++t) { px[t] = cA[2 * t]; py[t] = cA[2 * t + 1]; }
    for (int e = 0; e < 4; ++e) {
        float ex = cB[2 * e], ey = cB[2 * e + 1];
        int e1 = (e + 1) & 3;
        float dx = cB[2 * e1] - ex, dy = cB[2 * e1 + 1] - ey;
        float qx[16], qy[16]; int m = 0;
        for (int t = 0; t < n; ++t) {
            int t1 = (t + 1 == n) ? 0 : t + 1;
            float x0 = px[t], y0 = py[t], x1 = px[t1], y1 = py[t1];
            float s0 = dx * (y0 - ey) - dy * (x0 - ex);
            float s1 = dx * (y1 - ey) - dy * (x1 - ex);
            bool in0 = s0 >= 0.f, in1 = s1 >= 0.f;
            if (in0) { qx[m] = x0; qy[m] = y0; ++m; }
            if (in0 != in1) {
                float tt = s0 / (s0 - s1);
                qx[m] = x0 + tt * (x1 - x0); qy[m] = y0 + tt * (y1 - y0); ++m;
            }
        }
        n = m;
        for (int t = 0; t < n; ++t) { px[t] = qx[t]; py[t] = qy[t]; }
        if (n == 0) break;
    }
    float ar = 0.f;
    for (int t = 0; t < n; ++t) {
        int t1 = (t + 1 == n) ? 0 : t + 1;
        ar += px[t] * py[t1] - px[t1] * py[t];
    }
    return 0.5f * fabsf(ar);
}

// ---------------- K6: WMMA distance-matrix gate + exact IoU on survivors ----------------
// One wave per 16x16 tile of the 1024x1024 pair matrix (upper triangle).
// d^2(i,j) = n_i + n_j - 2 x_i x_j - 2 y_i y_j  via V_WMMA_F32_16X16X4_F32:
//   A row i = [x_i, y_i, n_i, 1] (16x4), B col j = [-2x_j, -2y_j, 1, n_j] (4x16).
__global__ __launch_bounds__(32) void k6_tiles(const float4* __restrict__ ab,
                                               const int* __restrict__ labels_k,
                                               const float* __restrict__ radius,
                                               const float* __restrict__ area,
                                               const float* __restrict__ corners,
                                               unsigned* __restrict__ mask) {
    int tile = blockIdx.x;
    int ti = tile >> 6, tj = tile & 63;
    if (tj < ti) return;           // wave-uniform early exit (before WMMA: EXEC all-ones where it runs)
    int i0 = ti << 4, j0 = tj << 4;
    int lane = threadIdx.x;
    int half = lane >> 4;          // 0: K=0,1   1: K=2,3  (32-bit A/B fragment layout)
    int l15  = lane & 15;

    float4 pa = ab[i0 + l15];
    float4 pb = ab[j0 + l15];
    v2f a, b;
    if (!half) { a.x = pa.x;  a.y = pa.y;  b.x = -2.0f * pb.x; b.y = -2.0f * pb.y; }
    else       { a.x = pa.z;  a.y = pa.w;  b.x = 1.0f;         b.y = pb.z;        }

    v8f c = {};
    c = __builtin_amdgcn_wmma_f32_16x16x4_f32(false, a, false, b, (short)0, c, false, false);

    int j  = j0 + l15;
    int lj = labels_k[j];
    float rj = radius[j];
#pragma unroll
    for (int v = 0; v < 8; ++v) {
        int i = i0 + v + (half << 3);   // C layout: VGPR v -> M=v (lanes 0-15) / M=v+8 (lanes 16-31), N=lane%16
        if (j <= i) continue;
        int li = labels_k[i];
        if (li < 0 || li != lj) continue;            // class-aware: offset boxes never overlap cross-class
        float rs = radius[i] + rj;
        float d2 = c[v];
        if (d2 > rs * rs) continue;                  // bounding-circle gate
        float inter = inter_area(&corners[i * 8], &corners[j * 8]);
        float u = inter / (area[i] + area[j] - inter + 1e-6f);
        if (u > NMS_THRESH)
            atomicOr(&mask[i * 32 + (j >> 5)], 1u << (j & 31));
    }
}

// ---------------- K7: sequential greedy suppression (single wave holds keep bits) ----------------
__global__ void k7_nms(const unsigned* __restrict__ mask, const float* __restrict__ vals,
                       unsigned* __restrict__ keepOut) {
    __shared__ unsigned rowAny[32];
    if (threadIdx.x < 32) rowAny[threadIdx.x] = 0u;
    __syncthreads();
    for (int r = threadIdx.x; r < TOPK_N; r += blockDim.x) {
        unsigned any = 0u;
        for (int w = 0; w < 32; ++w) any |= mask[r * 32 + w];
        if (any) atomicOr(&rowAny[r >> 5], 1u << (r & 31));
    }
    __syncthreads();
    if (threadIdx.x < 32) {
        int lane = threadIdx.x;
        unsigned keepW = 0u;
#pragma unroll
        for (int bIt = 0; bIt < 32; ++bIt) {
            int i = lane * 32 + bIt;
            if (i < TOPK_N && vals[i] > -1e30f) keepW |= 1u << bIt;
        }
        for (int i = 0; i < TOPK_N; ++i) {
            unsigned kw = __shfl(keepW, i >> 5);
            if (((kw >> (i & 31)) & 1u) && ((rowAny[i >> 5] >> (i & 31)) & 1u)) {
                unsigned m = mask[i * 32 + lane];   // bits only for j > i with IoU > thresh
                keepW &= ~m;
            }
        }
        keepOut[lane] = keepW;
    }
}

// ---------------- K8: top-300 of kept scores, write outputs ----------------
__global__ void k8_final(const float* __restrict__ vals, const unsigned* __restrict__ keep,
                         const float* __restrict__ boxes_k, const int* __restrict__ labels_k,
                         float* __restrict__ out) {
    __shared__ float s[1024];
    __shared__ int   id[1024];
    unsigned t = threadIdx.x;
    bool kb = (t < TOPK_N) && ((keep[t >> 5] >> (t & 31)) & 1u);
    s[t] = kb ? vals[t] : NEG_INF_F;
    id[t] = (int)t;
    __syncthreads();
    for (unsigned k = 2; k <= 1024; k <<= 1) {
        for (unsigned j = k >> 1; j > 0; j >>= 1) {
            unsigned ixj = t ^ j;
            if (ixj > t) {
                bool up = ((t & k) == 0);
                float sa = s[t], sb = s[ixj];
                int   ia = id[t], ib = id[ixj];
                bool before = (sa > sb) || (sa == sb && ia < ib);
                if (up ? !before : before) {
                    s[t] = sb; id[t] = ib; s[ixj] = sa; id[ixj] = ia;
                }
            }
            __syncthreads();
        }
    }
    if (t < DETS_PER_IMG) {
        int k = id[t];
        float sv = s[t];
        bool ok = sv > -1e30f;
#pragma unroll
        for (int q = 0; q < 5; ++q)
            out[t * 5 + q] = ok ? boxes_k[k * 5 + q] : 0.0f;
        out[1500 + t] = ok ? (float)labels_k[k] : -1.0f;
        out[1800 + t] = ok ? sv : 0.0f;
    }
}

// ---------------- host launch ----------------
extern "C" void kernel_launch(void* const* d_in, const int* in_sizes, int n_in,
                              void* d_out, int out_size, void* d_ws, size_t ws_size,
                              hipStream_t stream) {
    const float* boxes  = (const float*)d_in[0];
    const float* scores = (const float*)d_in[1];
    const int*   labels = (const int*)d_in[2];
    float* out = (float*)d_out;
    int n = in_sizes[1];

    char* ws = (char*)d_ws;
    unsigned* hist   = (unsigned*)(ws + OFF_HIST);
    unsigned* flags  = (unsigned*)(ws + OFF_FLAGS);
    float*    cscore = (float*)   (ws + OFF_CSCORE);
    unsigned* cidx   = (unsigned*)(ws + OFF_CIDX);
    float*    vals   = (float*)   (ws + OFF_VALS);
    int*      idx    = (int*)     (ws + OFF_IDX);
    float*    boxk   = (float*)   (ws + OFF_BOXK);
    int*      labk   = (int*)     (ws + OFF_LABK);
    float4*   ab     = (float4*)  (ws + OFF_AB);
    float*    corn   = (float*)   (ws + OFF_CORN);
    float*    rad    = (float*)   (ws + OFF_RAD);
    float*    area   = (float*)   (ws + OFF_AREA);
    unsigned* mask   = (unsigned*)(ws + OFF_MASK);
    unsigned* keep   = (unsigned*)(ws + OFF_KEEP);

    int zeroWords = (int)(WS_TOTAL / 4u);
    k0_zero<<<(zeroWords + 255) / 256, 256, 0, stream>>>((unsigned*)ws, zeroWords);
    k1_hist<<<1024, 256, 0, stream>>>(scores, n, hist);
    k2_scan<<<1, 1, 0, stream>>>(hist, flags);
    k3_compact<<<1024, 256, 0, stream>>>(scores, n, flags, &flags[2], cscore, cidx);
    k4_sort<<<1, 1024, 0, stream>>>(cscore, cidx, flags, vals, idx);
    k5_gather<<<4, 256, 0, stream>>>(boxes, labels, vals, idx, boxk, labk, ab, corn, rad, area);
    k6_tiles<<<4096, 32, 0, stream>>>(ab, labk, rad, area, corn, mask);
    k7_nms<<<1, 256, 0, stream>>>(mask, vals, keep);
    k8_final<<<1, 1024, 0, stream>>>(vals, keep, boxk, labk, out);
}